// NN_split_11647951307107
// MI455X (gfx1250) — compile-verified
//
#include <hip/hip_runtime.h>
#include <math.h>

#define B_SZ 256
#define NG   20000
#define G_SZ 4096
#define LMX  64

typedef __attribute__((ext_vector_type(2))) float v2f;
typedef __attribute__((ext_vector_type(8))) float v8f;

#define NWAVES  4
#define THREADS (NWAVES * 32)

// Grid: (G/16 group tiles, B/(16*NWAVES) batch quads). Each wave builds one
// 16x16 (b x g) hidden tile with V_WMMA_F32_16X16X4_F32 in block-diagonal
// form (exact f32 math). idx + masked W1 staged in LDS per block.
__global__ __launch_bounds__(THREADS) void group_linear_wmma(
    const float* __restrict__ x,
    const int*   __restrict__ group_idx,
    const int*   __restrict__ group_len,
    const float* __restrict__ W1,
    const float* __restrict__ b1,
    float*       __restrict__ hidden)
{
    __shared__ int   idx_lds[16 * LMX];   // 4 KB
    __shared__ float w1m_lds[16 * LMX];   // 4 KB (W1 * ragged mask)

    const int gt  = blockIdx.x;           // group tile, 0..255
    const int g0  = gt * 16;
    const int tid = threadIdx.x;

    // Stage this tile's indices and masked weights (coalesced, uniform flow).
    for (int e = tid; e < 16 * LMX; e += THREADS) {
        const int gi = e >> 6;
        const int l  = e & 63;
        const int g  = g0 + gi;
        idx_lds[e] = group_idx[g * LMX + l];
        const float w = W1[g * LMX + l];
        w1m_lds[e] = (l < group_len[g]) ? w : 0.0f;
    }
    __syncthreads();

    const int lane = tid & 31;
    const int wave = tid >> 5;
    const int half = lane >> 4;       // which 16-lane half
    const int m    = lane & 15;       // A row (b_local); C column (g_local)
    const int klo  = half * 2;        // K slots {0,1} vs {2,3} per A/B layout
    const int gcol = g0 + m;
    const int bt   = blockIdx.y * NWAVES + wave;   // this wave's batch tile
    const float bias = b1[gcol];
    const float* __restrict__ xrow = x + (size_t)(bt * 16 + m) * NG;

    v8f c0 = {};   // accumulator for even groups
    v8f c1 = {};   // accumulator for odd groups (WMMA ILP)

    for (int lc = 0; lc < LMX; lc += 4) {
        // Each lane's OWN group weights: for the single live lane (m==gi) of
        // any gi, w1m[gi*64+..] == w1m[m*64+..], so load unconditionally once
        // per K-chunk and select with cndmask per gi (no predicated LDS ops).
        const int wb = m * LMX + lc + klo;
        const float w0 = w1m_lds[wb];
        const float w1v = w1m_lds[wb + 1];

        #pragma unroll
        for (int gi = 0; gi < 16; gi += 2) {
            const int oa = gi * LMX + lc + klo;
            const int ob = oa + LMX;
            const int ia0 = idx_lds[oa];
            const int ia1 = idx_lds[oa + 1];
            const int ib0 = idx_lds[ob];
            const int ib1 = idx_lds[ob + 1];
            const bool selA = (m == gi);
            const bool selB = (m == gi + 1);
            v2f a, b, a2, b2;
            a.x  = xrow[ia0];
            a.y  = xrow[ia1];
            a2.x = xrow[ib0];
            a2.y = xrow[ib1];
            b.x  = selA ? w0  : 0.0f;
            b.y  = selA ? w1v : 0.0f;
            b2.x = selB ? w0  : 0.0f;
            b2.y = selB ? w1v : 0.0f;
            c0 = __builtin_amdgcn_wmma_f32_16x16x4_f32(
                     false, a,  false, b,  (short)0, c0, false, false);
            c1 = __builtin_amdgcn_wmma_f32_16x16x4_f32(
                     false, a2, false, b2, (short)0, c1, false, false);
        }
    }

    // Epilogue: bias + ReLU, store. C layout: lane = N(gcol), VGPR r = M.
    const int browbase = bt * 16 + half * 8;
    #pragma unroll
    for (int r = 0; r < 8; ++r) {
        float h = c0[r] + c1[r] + bias;
        h = fmaxf(h, 0.0f);
        hidden[(size_t)(browbase + r) * G_SZ + gcol] = h;
    }
}

// Head: logits = relu_hidden @ W2^T + b2, then 2-class log_softmax.
__global__ __launch_bounds__(256) void head_kernel(
    const float* __restrict__ hidden,
    const float* __restrict__ W2,
    const float* __restrict__ b2,
    float*       __restrict__ out)
{
    __shared__ float red0[256];
    __shared__ float red1[256];
    const int b = blockIdx.x;
    const int t = threadIdx.x;
    const float* __restrict__ hrow = hidden + (size_t)b * G_SZ;
    float s0 = 0.0f, s1 = 0.0f;
    for (int g = t; g < G_SZ; g += 256) {
        const float h = hrow[g];
        s0 += h * W2[g];
        s1 += h * W2[G_SZ + g];
    }
    red0[t] = s0; red1[t] = s1;
    __syncthreads();
    for (int s = 128; s > 0; s >>= 1) {
        if (t < s) { red0[t] += red0[t + s]; red1[t] += red1[t + s]; }
        __syncthreads();
    }
    if (t == 0) {
        const float l0 = red0[0] + b2[0];
        const float l1 = red1[0] + b2[1];
        const float mx = fmaxf(l0, l1);
        const float lse = mx + logf(expf(l0 - mx) + expf(l1 - mx));
        out[b * 2 + 0] = l0 - lse;
        out[b * 2 + 1] = l1 - lse;
    }
}

extern "C" void kernel_launch(void* const* d_in, const int* in_sizes, int n_in,
                              void* d_out, int out_size, void* d_ws, size_t ws_size,
                              hipStream_t stream) {
    const float* x         = (const float*)d_in[0];
    const int*   group_idx = (const int*)  d_in[1];
    const int*   group_len = (const int*)  d_in[2];
    const float* W1        = (const float*)d_in[3];
    const float* b1        = (const float*)d_in[4];
    const float* W2        = (const float*)d_in[5];
    const float* b2        = (const float*)d_in[6];
    float* out    = (float*)d_out;
    float* hidden = (float*)d_ws;   // 256*4096*4 = 4 MB scratch

    dim3 grid(G_SZ / 16, B_SZ / (16 * NWAVES));   // (256, 4)
    group_linear_wmma<<<grid, THREADS, 0, stream>>>(
        x, group_idx, group_len, W1, b1, hidden);
    head_kernel<<<dim3(B_SZ), 256, 0, stream>>>(hidden, W2, b2, out);
}